// PairedGenerativeLoss_23785528886175
// MI455X (gfx1250) — compile-verified
//
#include <hip/hip_runtime.h>
#include <math.h>

// GP paired generative loss for MI455X (gfx1250, wave32).
// P=16, N=256, B=8192, D=16. Dominant op: [B,N] @ [N,N] f32 GEMM done with
// V_WMMA_F32_16X16X4_F32 (f32 kept for numerical cancellation safety).

#define P_   16
#define N_   256
#define B_   8192
#define D_   16
#define BT   64          // B-rows per workgroup
#define LDA  260         // padded LDS row stride (floats) -> conflict-free frag loads
#define LOG_2PI_F 1.8378770664093453f
#define SQRT5_F   2.2360679774997896f
#define FIVE_THIRDS 1.6666666666666667f

typedef __attribute__((ext_vector_type(2))) float v2f;
typedef __attribute__((ext_vector_type(8))) float v8f;

__device__ __forceinline__ float matern52(float d, float d2) {
  // exp(-sqrt(5) d) * (sqrt(5) d + 1 + (5/3) d^2)
  return __expf(-SQRT5_F * d) * (SQRT5_F * d + 1.0f + FIVE_THIRDS * d2);
}

extern "C" __global__ __launch_bounds__(256)
void gp_pair_loss_kernel(const float* __restrict__ x1g, const float* __restrict__ x2g,
                         const float* __restrict__ targets,
                         const float* __restrict__ out_scales,
                         const float* __restrict__ noises,
                         const float* __restrict__ solved_means,
                         const float* __restrict__ chol,
                         const float* __restrict__ oxs,
                         float* __restrict__ ws)
{
  extern __shared__ float smem[];
  float* mat1 = smem;                    // BT*LDA
  float* mat2 = mat1 + BT * LDA;         // BT*LDA
  float* x1s  = mat2 + BT * LDA;         // BT*D
  float* x2s  = x1s  + BT * D_;          // BT*D
  float* x1sq = x2s  + BT * D_;          // BT
  float* x2sq = x1sq + BT;               // BT
  float* smS  = x2sq + BT;               // N
  float* s1p  = smS  + N_;               // 2*BT
  float* s2p  = s1p  + 2 * BT;           // 2*BT
  float* s12p = s2p  + 2 * BT;           // 2*BT
  float* mdp  = s12p + 2 * BT;           // 4*BT
  float* lossArr = mdp + 4 * BT;         // BT

  const int t    = threadIdx.x;
  const int p    = blockIdx.x >> 7;              // blockIdx / (B/BT) = /128
  const int rb64 = (blockIdx.x & 127) * BT;      // base row in B
  const float os    = out_scales[p];
  const float noise = noises[p];
  const float* cholP = chol + (size_t)p * N_ * N_;

  // ---- Phase 1a: stage x tiles (coalesced float4) + solved_means ----
  {
    const float4* x1v = (const float4*)(x1g + ((size_t)p * B_ + rb64) * D_);
    const float4* x2v = (const float4*)(x2g + ((size_t)p * B_ + rb64) * D_);
    ((float4*)x1s)[t] = x1v[t];          // 256 * float4 = BT*D floats
    ((float4*)x2s)[t] = x2v[t];
    smS[t] = solved_means[p * N_ + t];
  }
  __syncthreads();
  if (t < BT) {
    float a = 0.f, b = 0.f;
#pragma unroll
    for (int d = 0; d < D_; ++d) {
      float u = x1s[t * D_ + d]; a += u * u;
      float v = x2s[t * D_ + d]; b += v * v;
    }
    x1sq[t] = a; x2sq[t] = b;
  }
  // each thread owns inducing point n = t (16 floats in registers)
  float xr[D_];
  float y2 = 0.f;
  {
    const float* xp = oxs + ((size_t)p * N_ + t) * D_;
#pragma unroll
    for (int d = 0; d < D_; ++d) { xr[d] = xp[d]; y2 += xr[d] * xr[d]; }
  }
  __syncthreads();

  // ---- Phase 1b: matern tiles (x out_scale), row-major [BT][LDA] ----
  for (int r = 0; r < BT; ++r) {
    float dot1 = 0.f, dot2 = 0.f;
#pragma unroll
    for (int d = 0; d < D_; ++d) {
      dot1 += x1s[r * D_ + d] * xr[d];
      dot2 += x2s[r * D_ + d] * xr[d];
    }
    float d21 = fmaxf(x1sq[r] + y2 - 2.f * dot1, 0.f);
    float d22 = fmaxf(x2sq[r] + y2 - 2.f * dot2, 0.f);
    mat1[r * LDA + t] = matern52(sqrtf(d21), d21) * os;
    mat2[r * LDA + t] = matern52(sqrtf(d22), d22) * os;
  }
  __syncthreads();

  // ---- Phase 2: raw = matern @ chol via V_WMMA_F32_16X16X4_F32 ----
  // wave w: row block (w&3), column tiles cb with cb%2 == w>>2 (8 tiles).
  // chol is upper-triangular: column tile cb only needs K < 16*(cb+1).
  const int lane = t & 31;
  const int wv   = t >> 5;
  const int rbk  = wv & 3;
  const int cp   = wv >> 2;
  const int mRow = rbk * 16 + (lane & 15);
  const int kOff = (lane >> 4) << 1;     // lanes 0-15: K+0/1, lanes 16-31: K+2/3
  const int colL = lane & 15;

  const v8f vzero = {0.f,0.f,0.f,0.f,0.f,0.f,0.f,0.f};
  v8f sq1 = vzero, sq2 = vzero, s12 = vzero;

  for (int j = 0; j < 8; ++j) {
    const int cb = 2 * j + cp;
    const float* cB = cholP + cb * 16 + colL;
    const float* aRow1 = mat1 + mRow * LDA + kOff;
    const float* aRow2 = mat2 + mRow * LDA + kOff;
    v8f acc1 = vzero, acc2 = vzero;
    const int ksteps = cb * 4 + 4;       // triangular: K < 16*(cb+1)
    for (int ks = 0; ks < ksteps; ++ks) {
      const int k0 = ks * 4 + kOff;
      v2f bf;
      bf.x = cB[(size_t)k0 * N_];
      bf.y = cB[(size_t)(k0 + 1) * N_];
      v2f a1 = *(const v2f*)(aRow1 + ks * 4);
      v2f a2 = *(const v2f*)(aRow2 + ks * 4);
      acc1 = __builtin_amdgcn_wmma_f32_16x16x4_f32(false, a1, false, bf,
                                                   (short)0, acc1, false, false);
      acc2 = __builtin_amdgcn_wmma_f32_16x16x4_f32(false, a2, false, bf,
                                                   (short)0, acc2, false, false);
    }
    sq1 += acc1 * acc1;                  // sum raw^2 over N (elementwise, reduce later)
    sq2 += acc2 * acc2;
    s12 += acc1 * acc2;                  // r1 . r2
  }

  // reduce each C-element row across the 16 lanes of its half-wave
#pragma unroll
  for (int i = 0; i < 8; ++i) {
    float a = sq1[i], b = sq2[i], c = s12[i];
#pragma unroll
    for (int m = 1; m < 16; m <<= 1) {
      a += __shfl_xor(a, m, 32);
      b += __shfl_xor(b, m, 32);
      c += __shfl_xor(c, m, 32);
    }
    if ((lane & 15) == 0) {
      const int row = rbk * 16 + i + ((lane >> 4) << 3);   // M=i (lo half) / i+8 (hi half)
      s1p [cp * BT + row] = a;
      s2p [cp * BT + row] = b;
      s12p[cp * BT + row] = c;
    }
  }
  __syncthreads();

  // ---- Phase 3: means-diff GEMV (means[p] cancels in m1-m2) ----
  {
    const int row = t & 63, q = t >> 6;
    float acc = 0.f;
    const int n0 = q * 64;
#pragma unroll 8
    for (int n = n0; n < n0 + 64; ++n)
      acc += (mat1[row * LDA + n] - mat2[row * LDA + n]) * smS[n];
    mdp[q * BT + row] = acc;
  }
  __syncthreads();

  // ---- Phase 4: per-row pair-diff posterior + loss ----
  if (t < BT) {
    const int r = t;
    float dd2 = 0.f;
#pragma unroll
    for (int d = 0; d < D_; ++d) {
      float df = x1s[r * D_ + d] - x2s[r * D_ + d];
      dd2 += df * df;
    }
    const float rho = matern52(sqrtf(dd2), dd2) * os;
    const float c1  = os + noise - (s1p[r] + s1p[BT + r]);
    const float c2  = os + noise - (s2p[r] + s2p[BT + r]);
    const float dotv = s12p[r] + s12p[BT + r];
    const float dm   = dotv - rho;
    const float c1c2 = c1 * c2;
    const float rho2 = (fabsf(dm) >= c1c2) ? c1c2 : dm;
    const float cd   = fmaxf(c1 + c2 + 2.f * rho2, noise * noise);
    const float md   = mdp[r] + mdp[BT + r] + mdp[2 * BT + r] + mdp[3 * BT + r];
    const float tgt  = targets[(size_t)p * B_ + rb64 + r];
    const float e    = md - tgt;
    lossArr[r] = cd * e * e + __logf(cd) + LOG_2PI_F;
  }
  __syncthreads();
  if (t == 0) {                          // fixed-order (deterministic) block sum
    float s = 0.f;
    for (int r = 0; r < BT; ++r) s += lossArr[r];
    ws[blockIdx.x] = s;
  }
}

extern "C" __global__ __launch_bounds__(128)
void gp_finalize_kernel(const float* __restrict__ ws, float* __restrict__ out)
{
  __shared__ float red[128];
  const int p = blockIdx.x, t = threadIdx.x;
  red[t] = ws[p * 128 + t];
  __syncthreads();
#pragma unroll
  for (int s = 64; s > 0; s >>= 1) {     // fixed-order tree -> deterministic
    if (t < s) red[t] += red[t + s];
    __syncthreads();
  }
  if (t == 0) out[p] = red[0] * (0.5f / (float)B_);   // *0.5 then mean over B
}

extern "C" void kernel_launch(void* const* d_in, const int* in_sizes, int n_in,
                              void* d_out, int out_size, void* d_ws, size_t ws_size,
                              hipStream_t stream)
{
  const float* x1  = (const float*)d_in[0];   // inputs1 [P,B,D]
  const float* x2  = (const float*)d_in[1];   // inputs2 [P,B,D]
  const float* tg  = (const float*)d_in[2];   // targets [P,B]
  // d_in[3] = means [P]: cancels in means_diff, unused
  const float* os  = (const float*)d_in[4];   // out_scales [P]
  const float* no  = (const float*)d_in[5];   // noises [P]
  const float* smv = (const float*)d_in[6];   // solved_means [P,N]
  const float* ch  = (const float*)d_in[7];   // cholesky [P,N,N] (upper tri)
  const float* ox  = (const float*)d_in[8];   // original_xs [P,N,D]
  float* ws  = (float*)d_ws;
  float* out = (float*)d_out;

  const size_t shmem =
      (size_t)(2 * BT * LDA + 2 * BT * D_ + 2 * BT + N_ + 3 * 2 * BT + 4 * BT + BT)
      * sizeof(float);                                  // ~142 KB dynamic LDS

  hipLaunchKernelGGL(gp_pair_loss_kernel, dim3(P_ * (B_ / BT)), dim3(256), shmem, stream,
                     x1, x2, tg, os, no, smv, ch, ox, ws);
  hipLaunchKernelGGL(gp_finalize_kernel, dim3(P_), dim3(128), 0, stream, ws, out);
}